// GLRK4th_33758442947049
// MI455X (gfx1250) — compile-verified
//
#include <hip/hip_runtime.h>

typedef __attribute__((ext_vector_type(2))) float v2f;
typedef __attribute__((ext_vector_type(8))) float v8f;
typedef int v4i __attribute__((vector_size(16)));

#define LDSS 68     // padded LDS row stride (floats): rows 16B-aligned, conflict-free tile reads
#define NITER 24    // fixed-point iterations; spectral radius of h*M ~0.1 -> fp32-converged

#if __has_builtin(__builtin_amdgcn_global_load_async_to_lds_b128) && \
    __has_builtin(__builtin_amdgcn_global_store_async_from_lds_b128) && \
    __has_builtin(__builtin_amdgcn_s_wait_asynccnt)
#define USE_ASYNC 1
#else
#define USE_ASYNC 0
#endif

#define AS1 __attribute__((address_space(1)))
#define AS3 __attribute__((address_space(3)))

__device__ __forceinline__ AS1 v4i* as_global_v4i(const float* p) {
  AS1 const float* g = (AS1 const float*)p;   // addrspacecast generic -> global
  return (AS1 v4i*)g;                          // reinterpret pointee, drop const
}
__device__ __forceinline__ AS3 v4i* as_lds_v4i(const float* p) {
  AS3 const float* l = (AS3 const float*)p;   // addrspacecast generic -> LDS
  return (AS3 v4i*)l;                          // reinterpret pointee, drop const
}

__global__ __launch_bounds__(256) void glrk4_kernel(
    const float* __restrict__ gA1, const float* __restrict__ gA2,
    const float* __restrict__ gy0, const float* __restrict__ ghp,
    float* __restrict__ out_y, float* __restrict__ out_stack, int nB)
{
  __shared__ float sA1[64 * LDSS];
  __shared__ float sA2[64 * LDSS];
  __shared__ float sU[64], sWv[64], sK1[64], sK2[64], sR1[64], sR2[64], sY0[64];

  const int b = blockIdx.x;
  const int t = threadIdx.x;
  const float h = *ghp;

  const float c11 = 0.25f;
  const float c12 = 0.25f - 0.28867513459481287f;  // 0.25 - sqrt(3)/6
  const float c21 = 0.25f + 0.28867513459481287f;
  const float c22 = 0.25f;

  const float* A1b = gA1 + (size_t)b * 4096;
  const float* A2b = gA2 + (size_t)b * 4096;
  float* o1 = out_stack + (size_t)b * 4096;
  float* o2 = out_stack + (size_t)nB * 4096 + (size_t)b * 4096;

  // ---- Stage A1/A2 into LDS (padded layout). Async path: global->LDS, no VGPR round-trip.
  #pragma unroll
  for (int p = 0; p < 4; ++p) {
    const int idx = p * 1024 + t * 4;
    const int row = idx >> 6, col = idx & 63;
#if USE_ASYNC
    __builtin_amdgcn_global_load_async_to_lds_b128(
        as_global_v4i(A1b + idx), as_lds_v4i(sA1 + row * LDSS + col), 0, 0);
    __builtin_amdgcn_global_load_async_to_lds_b128(
        as_global_v4i(A2b + idx), as_lds_v4i(sA2 + row * LDSS + col), 0, 0);
#else
    float4 d1 = *(const float4*)(A1b + idx);
    float4 d2 = *(const float4*)(A2b + idx);
    *(float4*)(sA1 + row * LDSS + col) = d1;
    *(float4*)(sA2 + row * LDSS + col) = d2;
    *(float4*)(o1 + idx) = d1;
    *(float4*)(o2 + idx) = d2;
#endif
  }
  if (t < 64) sY0[t] = gy0[(size_t)b * 64 + t];
#if USE_ASYNC
  __builtin_amdgcn_s_wait_asynccnt(0);   // __syncthreads only covers DScnt, not ASYNCcnt
#endif
  __syncthreads();

#if USE_ASYNC
  // ---- Output copy: LDS -> global async stores; they drain while the solve runs.
  #pragma unroll
  for (int p = 0; p < 4; ++p) {
    const int idx = p * 1024 + t * 4;
    const int row = idx >> 6, col = idx & 63;
    __builtin_amdgcn_global_store_async_from_lds_b128(
        as_global_v4i(o1 + idx), as_lds_v4i(sA1 + row * LDSS + col), 0, 0);
    __builtin_amdgcn_global_store_async_from_lds_b128(
        as_global_v4i(o2 + idx), as_lds_v4i(sA2 + row * LDSS + col), 0, 0);
  }
#endif

  const int wave = t >> 5;          // 8 waves: 0-3 -> A1 row-tiles, 4-7 -> A2 row-tiles
  const int lane = t & 31;
  const int jt   = wave & 3;        // 16-row output tile
  const int cofs = (lane >> 4) * 2; // k sub-offset per half-wave (WMMA f32 operand layout)
  const float* sM   = (wave < 4) ? sA1 : sA2;
  const float* rowp = sM + (jt * 16 + (lane & 15)) * LDSS + cofs;

  // B operand: Mat^T tile per k-chunk. Load ONCE, then pin in VGPRs (opaque asm) so the
  // compiler cannot rematerialize the LDS loads inside the iteration loop.
  v2f Bop[16];
  #pragma unroll
  for (int c = 0; c < 16; ++c) {
    Bop[c].x = rowp[4 * c];
    Bop[c].y = rowp[4 * c + 1];
  }
  #pragma unroll
  for (int c = 0; c < 16; ++c) {
    asm("" : "+v"(Bop[c]));
  }

  // ---- R1 = A1 y0, R2 = A2 y0 (also first iterate K = R).
  {
    v8f acc = {};
    const float* vp = sY0 + cofs;
    #pragma unroll
    for (int c = 0; c < 16; ++c) {
      v2f a; a.x = vp[4 * c]; a.y = vp[4 * c + 1];   // broadcast vector chunk into A operand
      acc = __builtin_amdgcn_wmma_f32_16x16x4_f32(false, a, false, Bop[c],
                                                  (short)0, acc, false, false);
    }
    if (lane < 16) {
      int j = jt * 16 + lane;
      float r = acc[0];                               // D rows identical; row M=0 = lanes 0-15
      if (wave < 4) { sR1[j] = r; sK1[j] = r; }
      else          { sR2[j] = r; sK2[j] = r; }
    }
  }
  __syncthreads();

  // ---- Fixed-point: K <- R + h*M*K
  // k1' = R1 + A1*(h*(c11 k1 + c12 k2)),  k2' = R2 + A2*(h*(c21 k1 + c22 k2))
  for (int it = 0; it < NITER; ++it) {
    if (t < 64) {
      float k1v = sK1[t], k2v = sK2[t];
      sU[t]  = h * (c11 * k1v + c12 * k2v);
      sWv[t] = h * (c21 * k1v + c22 * k2v);
    }
    __syncthreads();
    v8f acc = {};
    const float* vp = ((wave < 4) ? sU : sWv) + cofs;
    #pragma unroll
    for (int c = 0; c < 16; ++c) {
      v2f a; a.x = vp[4 * c]; a.y = vp[4 * c + 1];
      acc = __builtin_amdgcn_wmma_f32_16x16x4_f32(false, a, false, Bop[c],
                                                  (short)0, acc, false, false);
    }
    if (lane < 16) {
      int j = jt * 16 + lane;
      if (wave < 4) sK1[j] = sR1[j] + acc[0];
      else          sK2[j] = sR2[j] + acc[0];
    }
    __syncthreads();
  }

  if (t < 64) {
    out_y[(size_t)b * 64 + t] = sY0[t] + h * 0.5f * (sK1[t] + sK2[t]);
  }
}

extern "C" void kernel_launch(void* const* d_in, const int* in_sizes, int n_in,
                              void* d_out, int out_size, void* d_ws, size_t ws_size,
                              hipStream_t stream) {
  (void)n_in; (void)out_size; (void)d_ws; (void)ws_size;
  const float* A1 = (const float*)d_in[0];
  const float* A2 = (const float*)d_in[1];
  const float* y0 = (const float*)d_in[2];
  const float* h  = (const float*)d_in[3];
  const int nB = in_sizes[0] / 4096;           // (B, 64, 64)
  float* out_y     = (float*)d_out;            // (B, 64)
  float* out_stack = out_y + (size_t)nB * 64;  // (2, B, 64, 64)
  glrk4_kernel<<<dim3(nB), dim3(256), 0, stream>>>(A1, A2, y0, h, out_y, out_stack, nB);
}